// Sim_vec_730144440388
// MI455X (gfx1250) — compile-verified
//
#include <hip/hip_runtime.h>

// ---------------------------------------------------------------------------
// Problem constants (from reference setup_inputs)
// ---------------------------------------------------------------------------
#define NI 16     // n_image
#define NC 32     // n_caption
#define NL 32     // max_words
#define NR 36     // n_regions
#define ND 1024   // embed_dim
#define EPS 1e-8f

typedef __attribute__((ext_vector_type(2))) float v2f;
typedef __attribute__((ext_vector_type(8))) float v8f;

// f32 WMMA: D(16x16,f32) = A(16x4,f32) * B(4x16,f32) + C
// A: lane(0-15)=M,K={0,1}; lane(16-31)=M,K={2,3}.  B: lane%16=N, vgpr/khalf=K.
// C/D: lane%16=N, vgpr v -> M = v + 8*(lane/16).
__device__ __forceinline__ v8f wmma_f32(v2f a, v2f b, v8f c) {
  return __builtin_amdgcn_wmma_f32_16x16x4_f32(
      false, a, false, b, (short)0, c, false, false);
}

// ---------------------------------------------------------------------------
// Kernel 1: Dlw rows.  One block per row i (D blocks x 256 threads).
// ---------------------------------------------------------------------------
__global__ __launch_bounds__(256) void dlw_kernel(const float* __restrict__ W,
                                                  const float* __restrict__ simw,
                                                  const float* __restrict__ tlw,
                                                  float* __restrict__ Dlw) {
  __shared__ float r1[256], r2[256];
  __shared__ float sh_thres, sh_tl;
  const int i = blockIdx.x;
  const int t = threadIdx.x;
  // idx = floor(linspace(0, D, D))[i] = floor(i*D/(D-1)), clipped
  int ridx = (i * ND) / (ND - 1);
  if (ridx > ND - 1) ridx = ND - 1;
  const float* Wr = W + (size_t)ridx * ND;

  float w[4];
  float s1 = 0.f, s2 = 0.f;
  for (int p = 0; p < 4; ++p) {
    float x = Wr[t + p * 256];
    float wv = 1.f / (1.f + expf(-x));
    w[p] = wv;
    s1 += wv;
    s2 += wv * wv;
  }
  r1[t] = s1; r2[t] = s2;
  __syncthreads();
  for (int s = 128; s > 0; s >>= 1) {
    if (t < s) { r1[t] += r1[t + s]; r2[t] += r2[t + s]; }
    __syncthreads();
  }
  if (t == 0) {
    float mean = r1[0] * (1.f / ND);
    float var = (r2[0] - (float)ND * mean * mean) / (float)(ND - 1); // ddof=1
    var = fmaxf(var, 0.f);
    sh_thres = mean + simw[i] * sqrtf(var);
    sh_tl = expf(tlw[0]);
  }
  __syncthreads();
  const float thres = sh_thres, tl = sh_tl;
  float m[4];
  float s3 = 0.f;
  for (int p = 0; p < 4; ++p) {
    float mask = tanhf(expf(tl * (w[p] - thres)));
    m[p] = mask * w[p];
    s3 += m[p] * m[p];
  }
  r1[t] = s3;
  __syncthreads();
  for (int s = 128; s > 0; s >>= 1) {
    if (t < s) r1[t] += r1[t + s];
    __syncthreads();
  }
  const float inv = 1.f / (sqrtf(r1[0]) + EPS);
  for (int p = 0; p < 4; ++p) Dlw[(size_t)i * ND + t + p * 256] = m[p] * inv;
}

// ---------------------------------------------------------------------------
// Kernel 2: diag[j] = sum_i Dlw[i][j]
// ---------------------------------------------------------------------------
__global__ void diag_kernel(const float* __restrict__ Dlw, float* __restrict__ diag) {
  const int j = blockIdx.x * blockDim.x + threadIdx.x;
  if (j < ND) {
    float s = 0.f;
    for (int r = 0; r < ND; ++r) s += Dlw[(size_t)r * ND + j];
    diag[j] = s;
  }
}

// ---------------------------------------------------------------------------
// Kernel 3: imgd[i,r,d] = img[i,r,d] * diag[d]
// ---------------------------------------------------------------------------
__global__ void imgd_kernel(const float* __restrict__ img,
                            const float* __restrict__ diag,
                            float* __restrict__ imgd) {
  const int idx = blockIdx.x * blockDim.x + threadIdx.x;
  if (idx < NI * NR * ND) imgd[idx] = img[idx] * diag[idx & (ND - 1)];
}

// ---------------------------------------------------------------------------
// Kernel 4: raw attn = tanh(cap @ imgd^T) -> LeakyReLU(0.1) -> word mask,
// stored layout (c,i,l,r).  One wave per 16x16 output tile.
// M = C*L = 1024 (64 tiles), N = I*R = 576 (36 tiles), K = D = 1024.
// ---------------------------------------------------------------------------
__global__ __launch_bounds__(32) void attn_raw_kernel(const float* __restrict__ cap,
                                                      const float* __restrict__ imgd,
                                                      const int* __restrict__ cap_lens,
                                                      float* __restrict__ attn) {
  const int m0 = blockIdx.x * 16;  // over C*L
  const int n0 = blockIdx.y * 16;  // over I*R
  const int lane = threadIdx.x;
  const int lo = lane & 15, hi = lane >> 4;
  const float* A = cap + (size_t)(m0 + lo) * ND;
  const float* B = imgd + (size_t)(n0 + lo) * ND;
  v8f acc = {};
  for (int k = 0; k < ND; k += 4) {
    const int kk = k + 2 * hi;
    v2f a = { A[kk], A[kk + 1] };
    v2f b = { B[kk], B[kk + 1] };
    acc = wmma_f32(a, b, acc);
  }
  for (int v = 0; v < 8; ++v) {
    const int m = m0 + v + 8 * hi;           // c*NL + l
    const int n = n0 + lo;                   // i*NR + r
    const int c = m >> 5, l = m & (NL - 1);
    const int i = n / NR, r = n % NR;
    float x = tanhf(acc[v]);
    x = (x >= 0.f) ? x : 0.1f * x;           // LeakyReLU(0.1)
    if (l >= cap_lens[c]) x = 0.f;           // zero padded words
    attn[(((size_t)c * NI + i) * NL + l) * NR + r] = x;
  }
}

// ---------------------------------------------------------------------------
// Kernel 5: fused per-(c,i) tail.  Block = (c,i), 256 threads = 8 waves.
//   - l2norm over l + softmax over r (in LDS)
//   - ctx = attn @ img_i           (WMMA, 16 rows at a time)
//   - sim_loc = l2norm(ctx)*cap    (64 KB LDS tile)
//   - num[l] = sim_loc . diag ;  sumsq[l] = || sim_loc @ Dlw^T ||^2  (WMMA,
//     sum-of-squares reduced via butterfly shuffles, output never stored)
//   - sim[c,i] = mean over valid words of num/(sqrt(sumsq)+eps)
// ---------------------------------------------------------------------------
__global__ __launch_bounds__(256) void fused_sim_kernel(
    const float* __restrict__ img, const float* __restrict__ cap,
    const float* __restrict__ diag, const float* __restrict__ Dlw,
    const float* __restrict__ attn, const int* __restrict__ cap_lens,
    const float* __restrict__ temp_scale, float* __restrict__ out) {
  __shared__ float s_attn[NL][NR];
  __shared__ float s_col[NR];
  __shared__ float s_simloc[16][ND];   // 64 KB (fits in 320 KB/WGP LDS)
  __shared__ float s_red[256];
  __shared__ float s_norm[16];
  __shared__ float s_num[16];
  __shared__ float s_wsum[8][16];
  __shared__ float s_simvec[NL];

  const int c = blockIdx.x, i = blockIdx.y;
  const int t = threadIdx.x;
  const int wave = t >> 5, lane = t & 31;
  const int lo = lane & 15, hi = lane >> 4;
  const float smooth = expf(temp_scale[0]);
  const float* imgI = img + (size_t)i * NR * ND;

  // ---- load raw attn tile ----
  for (int e = t; e < NL * NR; e += 256)
    s_attn[e / NR][e % NR] = attn[(((size_t)c * NI + i) * NL) * NR + e];
  __syncthreads();

  // ---- l2norm over words (axis l) per region column ----
  if (t < NR) {
    float s = 0.f;
    for (int l = 0; l < NL; ++l) { float v = s_attn[l][t]; s += v * v; }
    s_col[t] = sqrtf(s) + EPS;
  }
  __syncthreads();

  // ---- softmax over regions per word row ----
  if (t < NL) {
    float row[NR];
    float mx = -1e30f;
    for (int r = 0; r < NR; ++r) {
      float v = s_attn[t][r] / s_col[r] * smooth;
      row[r] = v;
      mx = fmaxf(mx, v);
    }
    float sum = 0.f;
    for (int r = 0; r < NR; ++r) { row[r] = expf(row[r] - mx); sum += row[r]; }
    const float inv = 1.f / sum;
    for (int r = 0; r < NR; ++r) s_attn[t][r] = row[r] * inv;
  }
  __syncthreads();

  for (int mh = 0; mh < 2; ++mh) {   // two 16-word halves
    // ---- ctx = attn(16xNR) @ imgI(NRxND); wave handles 8 n-tiles ----
    for (int j = 0; j < 8; ++j) {
      const int n0 = (wave * 8 + j) * 16;
      v8f acc = {};
      for (int k = 0; k < NR; k += 4) {          // NR = 36 = 9 * 4
        const int kk = k + 2 * hi;
        v2f a = { s_attn[mh * 16 + lo][kk], s_attn[mh * 16 + lo][kk + 1] };
        v2f b = { imgI[(size_t)kk * ND + n0 + lo],
                  imgI[(size_t)(kk + 1) * ND + n0 + lo] };
        acc = wmma_f32(a, b, acc);
      }
      for (int v = 0; v < 8; ++v)
        s_simloc[v + 8 * hi][n0 + lo] = acc[v];
    }
    __syncthreads();

    // ---- row l2 norms of ctx ----
    {
      const int l = t & 15, ch = t >> 4;
      float s = 0.f;
      for (int d = ch * 64; d < ch * 64 + 64; ++d) {
        float v = s_simloc[l][d];
        s += v * v;
      }
      s_red[t] = s;
    }
    __syncthreads();
    if (t < 16) {
      float s = 0.f;
      for (int ch = 0; ch < 16; ++ch) s += s_red[ch * 16 + t];
      s_norm[t] = sqrtf(s) + EPS;
    }
    __syncthreads();

    // ---- sim_loc = ctx/norm * cap ; partial num = sim_loc . diag ----
    {
      const int l = t & 15, ch = t >> 4;
      const float* capRow = cap + ((size_t)c * NL + mh * 16 + l) * ND;
      const float inv = 1.f / s_norm[l];
      float np = 0.f;
      for (int d = ch * 64; d < ch * 64 + 64; ++d) {
        float v = s_simloc[l][d] * inv * capRow[d];
        s_simloc[l][d] = v;
        np += v * diag[d];
      }
      s_red[t] = np;
    }
    __syncthreads();
    if (t < 16) {
      float s = 0.f;
      for (int ch = 0; ch < 16; ++ch) s += s_red[ch * 16 + t];
      s_num[t] = s;
    }
    __syncthreads();

    // ---- sumsq[l] of sim_loc @ Dlw^T ; wave handles 8 e-tiles ----
    float rs[8];
    for (int v = 0; v < 8; ++v) rs[v] = 0.f;
    for (int j = 0; j < 8; ++j) {
      const int e0 = (wave * 8 + j) * 16;
      const float* Brow = Dlw + (size_t)(e0 + lo) * ND;
      v8f acc = {};
      for (int k = 0; k < ND; k += 4) {
        const int kk = k + 2 * hi;
        v2f a = { s_simloc[lo][kk], s_simloc[lo][kk + 1] };
        v2f b = { Brow[kk], Brow[kk + 1] };
        acc = wmma_f32(a, b, acc);
      }
      for (int v = 0; v < 8; ++v) {
        float val = acc[v] * acc[v];
        val += __shfl_xor(val, 1, 32);   // butterfly over the 16-lane group
        val += __shfl_xor(val, 2, 32);
        val += __shfl_xor(val, 4, 32);
        val += __shfl_xor(val, 8, 32);
        rs[v] += val;                    // row m = v + 8*hi summed over e-tile
      }
    }
    if (lo == 0) {
      for (int v = 0; v < 8; ++v) s_wsum[wave][v + 8 * hi] = rs[v];
    }
    __syncthreads();
    if (t < 16) {
      float s = 0.f;
      for (int w = 0; w < 8; ++w) s += s_wsum[w][t];
      s_simvec[mh * 16 + t] = s_num[t] / (sqrtf(s) + EPS);
    }
    __syncthreads();
  }

  if (t == 0) {
    const int len = cap_lens[c];
    float s = 0.f;
    for (int l = 0; l < NL; ++l)
      if (l < len) s += s_simvec[l];
    out[(size_t)i * NC + c] = s / (float)len;   // sim.T -> (I, C)
  }
}

// ---------------------------------------------------------------------------
// Launch
// ---------------------------------------------------------------------------
extern "C" void kernel_launch(void* const* d_in, const int* in_sizes, int n_in,
                              void* d_out, int out_size, void* d_ws, size_t ws_size,
                              hipStream_t stream) {
  (void)in_sizes; (void)n_in; (void)out_size; (void)ws_size;
  const float* img  = (const float*)d_in[0];  // (I,R,D)
  const float* cap  = (const float*)d_in[1];  // (C,L,D)
  const float* W    = (const float*)d_in[2];  // (D,D)
  const float* simw = (const float*)d_in[3];  // (1,D)
  const float* tsc  = (const float*)d_in[4];  // (1,1) temp_scale
  const float* tlw  = (const float*)d_in[5];  // (1,1) temp_learnable
  const int*   lens = (const int*)d_in[6];    // (C,)
  float* out = (float*)d_out;

  float* ws   = (float*)d_ws;
  float* Dlw  = ws;                            // D*D
  float* diag = Dlw + (size_t)ND * ND;         // D
  float* imgd = diag + ND;                     // I*R*D
  float* attn = imgd + (size_t)NI * NR * ND;   // C*I*L*R

  dlw_kernel<<<ND, 256, 0, stream>>>(W, simw, tlw, Dlw);
  diag_kernel<<<(ND + 255) / 256, 256, 0, stream>>>(Dlw, diag);
  imgd_kernel<<<(NI * NR * ND + 255) / 256, 256, 0, stream>>>(img, diag, imgd);
  attn_raw_kernel<<<dim3((NC * NL) / 16, (NI * NR) / 16), 32, 0, stream>>>(
      cap, imgd, lens, attn);
  fused_sim_kernel<<<dim3(NC, NI), 256, 0, stream>>>(
      img, cap, diag, Dlw, attn, lens, tsc, out);
}